// DSFD_74655121539694
// MI455X (gfx1250) — compile-verified
//
#include <hip/hip_runtime.h>
#include <stdint.h>

// Problem constants (match reference)
#define NB      32
#define NP      34125
#define NK      1000
#define KW      32          // ceil(NK/32) keep/suppression words (u32)
#define NKP     1008        // NK padded to multiple of 16 for WMMA tiles
#define NT1     63          // NKP/16 tiles per dimension
#define NTILES  (NT1*NT1)
#define NBINS   4096
#define CAP     4096        // candidate capacity for per-image sort
#define VV0     0.1f
#define VV1     0.2f
#define NMS_THR 0.3f
#define CONF_THR 0.05f

// ---------------- workspace layout ----------------
__host__ __device__ constexpr size_t alup(size_t x) { return (x + 255) & ~(size_t)255; }
constexpr size_t O_SCORES = 0;                                          // NB*NP f32
constexpr size_t O_HIST   = alup(O_SCORES + (size_t)NB * NP * 4);       // NB*NBINS u32
constexpr size_t O_THR    = alup(O_HIST   + (size_t)NB * NBINS * 4);    // NB i32
constexpr size_t O_TOPS   = alup(O_THR    + (size_t)NB * 4);            // NB*NK f32
constexpr size_t O_TOPB   = alup(O_TOPS   + (size_t)NB * NK * 4);       // NB*NK*4 f32
constexpr size_t O_MASK   = alup(O_TOPB   + (size_t)NB * NK * 4 * 4);   // NB*NK*KW u32

// ---------------- CDNA5 async global->LDS helpers ----------------
typedef int   v4i __attribute__((ext_vector_type(4)));
typedef float v2f __attribute__((ext_vector_type(2)));
typedef float v8f __attribute__((ext_vector_type(8)));

__device__ __forceinline__ void async_copy16(void* lds_dst, const void* gsrc) {
#if __has_builtin(__builtin_amdgcn_global_load_async_to_lds_b128)
  // tracked with ASYNCcnt (CDNA5 ISA)
  __builtin_amdgcn_global_load_async_to_lds_b128(
      (__attribute__((address_space(1))) v4i*)(uintptr_t)gsrc,
      (__attribute__((address_space(3))) v4i*)(unsigned)(uintptr_t)lds_dst,
      0, 0);
#else
  *(uint4*)lds_dst = *(const uint4*)gsrc;   // sync fallback
#endif
}

__device__ __forceinline__ void wait_async0() {
#if __has_builtin(__builtin_amdgcn_s_wait_asynccnt)
  __builtin_amdgcn_s_wait_asynccnt(0);
#else
  asm volatile("s_wait_asynccnt 0" ::: "memory");
#endif
}

// ---------------- kernel 0: zero histogram ----------------
__global__ void dsfd_zero_hist(unsigned* __restrict__ hist) {
  int i = blockIdx.x * blockDim.x + threadIdx.x;
  if (i < NB * NBINS) hist[i] = 0u;
}

// ---------------- kernel 1: scores + per-image histogram ----------------
__global__ void dsfd_score_hist(const float* __restrict__ conf,
                                float* __restrict__ scores,
                                unsigned* __restrict__ hist) {
  int i = blockIdx.x * blockDim.x + threadIdx.x;
  if (i >= NB * NP) return;
  int b  = i / NP;
  float c0 = conf[(size_t)i * 2 + 0];
  float c1 = conf[(size_t)i * 2 + 1];
  float s  = 1.0f / (1.0f + expf(c0 - c1));   // softmax class-1 of 2 classes
  scores[i] = s;
  int bin = (int)(s * (float)NBINS);
  if (bin > NBINS - 1) bin = NBINS - 1;
  if (bin < 0) bin = 0;
  atomicAdd(&hist[b * NBINS + bin], 1u);
}

// ---------------- kernel 2: per-image score threshold bin ----------------
__global__ void dsfd_find_thr(const unsigned* __restrict__ hist,
                              int* __restrict__ thrbin) {
  int b = threadIdx.x;
  if (b >= NB) return;
  unsigned cum = 0;
  int thr = 0;
  for (int bin = NBINS - 1; bin >= 0; --bin) {
    cum += hist[b * NBINS + bin];
    if (cum >= (unsigned)NK) { thr = bin; break; }
  }
  thrbin[b] = thr;
}

// ---------------- kernel 3: gather candidates, bitonic top-K, decode boxes ----------------
__global__ __launch_bounds__(1024) void dsfd_topk(const float* __restrict__ scores,
                                                  const int* __restrict__ thrbin,
                                                  const float* __restrict__ loc,
                                                  const float* __restrict__ priors,
                                                  float* __restrict__ topS,
                                                  float* __restrict__ topB) {
  __shared__ unsigned long long keys[CAP];   // 32 KB
  __shared__ unsigned cnt;
  const int b = blockIdx.x, tid = threadIdx.x;
  if (tid == 0) cnt = 0u;
  for (int t = tid; t < CAP; t += 1024) keys[t] = 0ull;
  __syncthreads();

  const int thr = thrbin[b];
  for (int p = tid; p < NP; p += 1024) {
    float s = scores[(size_t)b * NP + p];
    int bin = (int)(s * (float)NBINS);
    if (bin > NBINS - 1) bin = NBINS - 1;
    if (bin >= thr) {
      unsigned pos = atomicAdd(&cnt, 1u);
      if (pos < CAP)
        keys[pos] = ((unsigned long long)__float_as_uint(s) << 32) |
                    (unsigned long long)(0xFFFFFFFFu - (unsigned)p);
    }
  }
  __syncthreads();

  // bitonic sort, descending: keys[0] = best (score desc, index asc on ties)
  for (unsigned ksz = 2; ksz <= CAP; ksz <<= 1) {
    for (unsigned j = ksz >> 1; j > 0; j >>= 1) {
      for (unsigned idx = tid; idx < CAP; idx += 1024) {
        unsigned ixj = idx ^ j;
        if (ixj > idx) {
          unsigned long long a = keys[idx], c = keys[ixj];
          bool desc = ((idx & ksz) == 0);
          if (desc ? (a < c) : (a > c)) { keys[idx] = c; keys[ixj] = a; }
        }
      }
      __syncthreads();
    }
  }

  // decode only the surviving top-K
  for (int k = tid; k < NK; k += 1024) {
    unsigned long long key = keys[k];
    float s = __uint_as_float((unsigned)(key >> 32));
    unsigned p = 0xFFFFFFFFu - (unsigned)(key & 0xFFFFFFFFull);
    size_t o = (size_t)b * NK + k;
    if (p >= (unsigned)NP) {            // padding (shouldn't happen): zero entry
      topS[o] = 0.0f;
      topB[o * 4 + 0] = topB[o * 4 + 1] = topB[o * 4 + 2] = topB[o * 4 + 3] = 0.0f;
      continue;
    }
    float l0 = loc[((size_t)b * NP + p) * 4 + 0];
    float l1 = loc[((size_t)b * NP + p) * 4 + 1];
    float l2 = loc[((size_t)b * NP + p) * 4 + 2];
    float l3 = loc[((size_t)b * NP + p) * 4 + 3];
    float p0 = priors[(size_t)p * 4 + 0], p1 = priors[(size_t)p * 4 + 1];
    float p2 = priors[(size_t)p * 4 + 2], p3 = priors[(size_t)p * 4 + 3];
    float cx = p0 + l0 * VV0 * p2;
    float cy = p1 + l1 * VV0 * p3;
    float w  = p2 * expf(l2 * VV1);
    float h  = p3 * expf(l3 * VV1);
    topS[o] = s;
    topB[o * 4 + 0] = cx - 0.5f * w;
    topB[o * 4 + 1] = cy - 0.5f * h;
    topB[o * 4 + 2] = cx + 0.5f * w;
    topB[o * 4 + 3] = cy + 0.5f * h;
  }
}

// ---------------- kernel 4: suppression bitmask via WMMA area-sum tiles ----------------
// Pair-union base Asum[i][j] = area_i + area_j computed on the matrix pipe:
//   D = A x B, A(16x4) rows = [area_i, 1, 0, 0], B(4x16) cols = [1; area_j; 0; 0]
// One wave per 16x16 (i,j) tile; intersection min/max on VALU; __ballot -> bits.
__global__ __launch_bounds__(256) void dsfd_supmask(const float* __restrict__ topB,
                                                    unsigned short* __restrict__ mask16) {
  __shared__ float4 sbox[NKP];              // ~16 KB, async-staged
  __shared__ float  sarea[NKP];             // ~4 KB
  const int b = blockIdx.y, tid = threadIdx.x;

  for (int t = tid; t < NK; t += 256)
    async_copy16(&sbox[t], topB + ((size_t)b * NK + t) * 4);
  for (int t = NK + tid; t < NKP; t += 256)
    sbox[t] = make_float4(0.f, 0.f, 0.f, 0.f);
  wait_async0();
  __syncthreads();

  for (int t = tid; t < NKP; t += 256) {
    float4 bx = sbox[t];
    sarea[t] = fmaxf(bx.z - bx.x, 0.0f) * fmaxf(bx.w - bx.y, 0.0f);
  }
  __syncthreads();

  const int lane = tid & 31, wave = tid >> 5;
  const int wid = blockIdx.x * 8 + wave;    // tile id (wave-uniform)
  if (wid >= NTILES) return;                // whole-wave exit: EXEC stays all-1s below
  const int it = wid / NT1, jt = wid % NT1;
  const int i0 = it * 16, j0 = jt * 16;
  const int half = lane >> 4, N = lane & 15;

  // A: lanes 0-15 hold K=0(V0)=area_i, K=1(V1)=1 ; lanes 16-31 hold K=2,3 = 0
  // B: lanes 0-15 hold K=0(V0)=1, K=1(V1)=area_j ; lanes 16-31 hold K=2,3 = 0
  v2f A, Bm;
  A.x  = half ? 0.0f : sarea[i0 + N];
  A.y  = half ? 0.0f : 1.0f;
  Bm.x = half ? 0.0f : 1.0f;
  Bm.y = half ? 0.0f : sarea[j0 + N];
  v8f C = {0.f, 0.f, 0.f, 0.f, 0.f, 0.f, 0.f, 0.f};
  C = __builtin_amdgcn_wmma_f32_16x16x4_f32(false, A, false, Bm,
                                            (short)0, C, false, false);
  // D layout: lane l, VGPR r -> row i0 + r + 8*(l>=16), col j0 + (l&15)
  const int j = j0 + N;
  const float4 bj = sbox[j];
  #pragma unroll
  for (int r = 0; r < 8; ++r) {
    int i = i0 + r + half * 8;
    float4 bi = sbox[i];
    float lx = fmaxf(bi.x, bj.x), ly = fmaxf(bi.y, bj.y);
    float rx = fminf(bi.z, bj.z), ry = fminf(bi.w, bj.w);
    float inter = fmaxf(rx - lx, 0.0f) * fmaxf(ry - ly, 0.0f);
    float iou = inter / (C[r] - inter + 1e-9f);     // C[r] = area_i + area_j (WMMA)
    bool pred = (i < NK) & (j < NK) & (j > i) & (iou > NMS_THR);
    unsigned bal = (unsigned)__ballot(pred);        // wave32: two 16-col rows
    if (lane == 0 && (i0 + r) < NK)
      mask16[((size_t)b * NK + (i0 + r)) * (KW * 2) + jt] = (unsigned short)(bal & 0xFFFFu);
    if (lane == 16 && (i0 + r + 8) < NK)
      mask16[((size_t)b * NK + (i0 + r + 8)) * (KW * 2) + jt] = (unsigned short)(bal >> 16);
  }
  // col tile 63 (cols 1008..1023) is never produced: zero it so u32 word 31 is valid
  if (jt == NT1 - 1 && half == 0) {
    int row = i0 + N;
    if (row < NK) mask16[((size_t)b * NK + row) * (KW * 2) + 63] = 0;
  }
}

// ---------------- kernel 5: wave32 greedy NMS scan + writeout ----------------
__global__ __launch_bounds__(128) void dsfd_nms_write(const float* __restrict__ topS,
                                                      const float* __restrict__ topB,
                                                      const unsigned* __restrict__ mask,
                                                      float* __restrict__ out) {
  extern __shared__ char smem[];            // NK*KW*4 = 128 KB (needs CDNA5 320KB LDS)
  __shared__ unsigned keepArr[KW];
  unsigned* lmask = (unsigned*)smem;
  const int b = blockIdx.x, tid = threadIdx.x;

  // async-stage the whole 128 KB suppression tile for this image
  const char* gmask = (const char*)(mask + (size_t)b * NK * KW);
  for (int t = tid; t < (NK * KW) / 4; t += 128)
    async_copy16(smem + (size_t)t * 16, gmask + (size_t)t * 16);
  wait_async0();
  __syncthreads();

  if (tid < 32) {                           // wave 0 does the sequential greedy scan
    const int lane = tid;
    unsigned kw_ = 0u;                      // keep word `lane`: init = score > CONF_THR
    for (int t = 0; t < 32; ++t) {
      int i = lane * 32 + t;
      if (i < NK && topS[(size_t)b * NK + i] > CONF_THR) kw_ |= (1u << t);
    }
    for (int i = 0; i < NK; ++i) {
      unsigned wrd = __shfl(kw_, i >> 5, 32);
      if ((wrd >> (i & 31)) & 1u)           // box i still kept -> suppress overlaps j>i
        kw_ &= ~lmask[i * KW + lane];
    }
    keepArr[lane] = kw_;
  }
  __syncthreads();

  for (int k = tid; k < NK; k += 128) {
    float f = (float)((keepArr[k >> 5] >> (k & 31)) & 1u);
    size_t o = (size_t)b * NK + k;
    out[o * 5 + 0] = topS[o] * f;
    out[o * 5 + 1] = topB[o * 4 + 0] * f;
    out[o * 5 + 2] = topB[o * 4 + 1] * f;
    out[o * 5 + 3] = topB[o * 4 + 2] * f;
    out[o * 5 + 4] = topB[o * 4 + 3] * f;
  }
}

// ---------------- launcher ----------------
extern "C" void kernel_launch(void* const* d_in, const int* in_sizes, int n_in,
                              void* d_out, int out_size, void* d_ws, size_t ws_size,
                              hipStream_t stream) {
  (void)in_sizes; (void)n_in; (void)out_size; (void)ws_size;
  const float* loc    = (const float*)d_in[0];   // [NB, NP, 4]
  const float* conf   = (const float*)d_in[1];   // [NB, NP, 2]
  const float* priors = (const float*)d_in[2];   // [NP, 4]
  float* out = (float*)d_out;                    // [NB, NK, 5]

  char* ws = (char*)d_ws;
  float*    scores = (float*)(ws + O_SCORES);
  unsigned* hist   = (unsigned*)(ws + O_HIST);
  int*      thrbin = (int*)(ws + O_THR);
  float*    topS   = (float*)(ws + O_TOPS);
  float*    topB   = (float*)(ws + O_TOPB);
  unsigned* mask   = (unsigned*)(ws + O_MASK);

  dsfd_zero_hist <<<(NB * NBINS + 255) / 256, 256, 0, stream>>>(hist);
  dsfd_score_hist<<<(NB * NP + 255) / 256,    256, 0, stream>>>(conf, scores, hist);
  dsfd_find_thr  <<<1, 32, 0, stream>>>(hist, thrbin);
  dsfd_topk      <<<NB, 1024, 0, stream>>>(scores, thrbin, loc, priors, topS, topB);
  dsfd_supmask   <<<dim3((NTILES + 7) / 8, NB), 256, 0, stream>>>(topB, (unsigned short*)mask);
  dsfd_nms_write <<<NB, 128, NK * KW * 4, stream>>>(topS, topB, mask, out);
}